// ARENALoRALayer_43310450213238
// MI455X (gfx1250) — compile-verified
//
#include <hip/hip_runtime.h>

// LoRA fused forward for MI455X (gfx1250, wave32, WMMA f32 16x16x4).
//   h = x @ V^T            [8192,4096] x [4096,32]
//   h *= S                 (per-rank scale)
//   out = h @ U^T          [8192,32] x [32,4096]   (SCALING = 32/32 = 1.0)
//
// One fused pass: x read once (128MB), out written once (128MB); U,V stay in L2.

typedef float v2f __attribute__((ext_vector_type(2)));
typedef float v8f __attribute__((ext_vector_type(8)));

#define IN_F   4096
#define OUT_F  4096
#define RANK   32
#define WAVES  8
#define BLOCK  (WAVES * 32)

__global__ __launch_bounds__(BLOCK) void lora_fused_wmma(
    const float* __restrict__ x,   // [8192, 4096]
    const float* __restrict__ U,   // [4096, 32]
    const float* __restrict__ S,   // [32]
    const float* __restrict__ V,   // [32, 4096]
    float* __restrict__ out)       // [8192, 4096]
{
    __shared__ __align__(16) float hPart[WAVES * 16 * RANK]; // 16 KB partials
    __shared__ __align__(16) float hFin[16 * RANK];          // 2 KB reduced h

    const int tid  = threadIdx.x;
    const int wave = tid >> 5;
    const int lane = tid & 31;
    const int l16  = lane & 15;      // M (A/D) or N (B) index within tile
    const int hi   = lane >> 4;      // half-wave selects K+0/1 vs K+2/3
    const int koff = hi * 2;
    const int m0   = blockIdx.x * 16;

    // ---------------- Phase 1: partial h[16,32] over this wave's K-slice ----
    v8f acc0 = {};   // rank 0..15
    v8f acc1 = {};   // rank 16..31
    {
        const int kw0 = wave * (IN_F / WAVES);        // 512-wide K slice
        const float* xrow  = x + (size_t)(m0 + l16) * IN_F + kw0 + koff;
        const float* v0row = V + (size_t)(l16)      * IN_F + kw0 + koff;
        const float* v1row = V + (size_t)(16 + l16) * IN_F + kw0 + koff;
        #pragma unroll 4
        for (int kk = 0; kk < IN_F / WAVES; kk += 4) {
            v2f a  = *(const v2f*)(xrow  + kk);   // A: x[m, k..k+1]
            v2f b0 = *(const v2f*)(v0row + kk);   // B: V[r, k..k+1], r = l16
            v2f b1 = *(const v2f*)(v1row + kk);   // B: V[16+r, k..k+1]
            acc0 = __builtin_amdgcn_wmma_f32_16x16x4_f32(
                       false, a, false, b0, (short)0, acc0, false, false);
            acc1 = __builtin_amdgcn_wmma_f32_16x16x4_f32(
                       false, a, false, b1, (short)0, acc1, false, false);
        }
    }

    // D layout: VGPR i -> (M = i + 8*hi, N = l16). Spill partials to LDS.
    {
        float* p = hPart + wave * (16 * RANK);
        #pragma unroll
        for (int i = 0; i < 8; ++i) {
            int m = i + 8 * hi;
            p[m * RANK + l16]      = acc0[i];
            p[m * RANK + 16 + l16] = acc1[i];
        }
    }
    __syncthreads();

    // ---------------- Cross-wave reduction + S scale (256 thr x float2) -----
    {
        v2f sum = {};
        #pragma unroll
        for (int w = 0; w < WAVES; ++w)
            sum += *(const v2f*)(hPart + w * (16 * RANK) + tid * 2);
        v2f sv = *(const v2f*)(S + 2 * (tid & 15));   // r = 2*(tid%16)+{0,1}
        sum *= sv;                                     // SCALING==1.0 folded
        *(v2f*)(hFin + tid * 2) = sum;
    }
    __syncthreads();

    // ---------------- Phase 2: out[16, n0..n0+511] = h @ U^T ----------------
    // Hoist A operand (h, K=32 -> 8 k-steps) once; reuse for all 32 N-tiles.
    v2f ah[8];
    #pragma unroll
    for (int kk = 0; kk < 8; ++kk)
        ah[kk] = *(const v2f*)(hFin + l16 * RANK + kk * 4 + koff);

    const int n0 = wave * (OUT_F / WAVES);            // 512-wide N slice
    for (int nt = 0; nt < 32; ++nt) {
        const int nc = n0 + nt * 16;
        const float* urow = U + (size_t)(nc + l16) * RANK + koff; // B: U[n, r..r+1]
        v8f acc = {};
        #pragma unroll
        for (int kk = 0; kk < 8; ++kk) {
            v2f b = *(const v2f*)(urow + kk * 4);
            acc = __builtin_amdgcn_wmma_f32_16x16x4_f32(
                      false, ah[kk], false, b, (short)0, acc, false, false);
        }
        // D layout: VGPR i -> row m0 + i + 8*hi, col nc + l16
        float* orow = out + (size_t)(m0 + 8 * hi) * OUT_F + nc + l16;
        #pragma unroll
        for (int i = 0; i < 8; ++i)
            orow[(size_t)i * OUT_F] = acc[i];
    }
}

extern "C" void kernel_launch(void* const* d_in, const int* in_sizes, int n_in,
                              void* d_out, int out_size, void* d_ws, size_t ws_size,
                              hipStream_t stream) {
    const float* x = (const float*)d_in[0];   // [4,2048,4096] -> [8192,4096]
    const float* U = (const float*)d_in[1];   // [4096,32]
    const float* S = (const float*)d_in[2];   // [32]
    const float* V = (const float*)d_in[3];   // [32,4096]
    float* out = (float*)d_out;               // [8192,4096]

    const int M = 4 * 2048;                   // 8192 rows
    dim3 grid(M / 16), block(BLOCK);
    lora_fused_wmma<<<grid, block, 0, stream>>>(x, U, S, V, out);
}